// PNG_LSTM_23493471109323
// MI455X (gfx1250) — compile-verified
//
#include <hip/hip_runtime.h>
#include <math.h>

typedef __attribute__((ext_vector_type(2))) float v2f;
typedef __attribute__((ext_vector_type(8))) float v8f;

#define B_SZ   4096
#define T_SZ   1024
#define I_SZ   8
#define H_SZ   12
#define G_SZ   48     // 4*H
#define NA     4

#define WPB    2      // waves per block
#define ROWS   16     // batch rows per wave (WMMA M)
#define PF_T   16     // prefetch distance in timesteps (16*8 floats = 512B ahead)

__device__ __forceinline__ float fast_sigmoid(float v) {
    // 1/(1+exp(-v)) via v_exp_f32 + v_rcp_f32
    return __builtin_amdgcn_rcpf(1.0f + __expf(-v));
}
__device__ __forceinline__ float fast_tanh(float v) {
    return 2.0f * fast_sigmoid(2.0f * v) - 1.0f;
}

__global__ __launch_bounds__(WPB * 32) void lstm_wmma_kernel(
    const float* __restrict__ x,      // [B, T, I]
    const float* __restrict__ W_ih,   // [48, 8]
    const float* __restrict__ W_hh,   // [48, 12]
    const float* __restrict__ b_ih,   // [48]
    const float* __restrict__ b_hh,   // [48]
    const float* __restrict__ W_fc,   // [4, 12]
    const float* __restrict__ b_fc,   // [4]
    float* __restrict__ out,          // [B, T, 4]
    float* __restrict__ hn,           // [B, 12]
    float* __restrict__ cn)           // [B, 12]
{
    __shared__ float gatesS[WPB][ROWS][G_SZ];
    __shared__ float hS[WPB][ROWS][H_SZ];
    __shared__ float cS[WPB][ROWS][H_SZ];

    const int tid   = threadIdx.x;
    const int wid   = tid >> 5;
    const int lane  = tid & 31;
    const int hw    = (lane < 16) ? 0 : 1;   // half-wave
    const int col16 = lane & 15;             // N / M index within tile
    const int b0    = (blockIdx.x * WPB + wid) * ROWS;

    // ---- Preload B-matrix fragments (constant over T) ----
    // B frag for K-chunk kc: VGPR0 holds row K=4*kc+2*hw, VGPR1 row K=4*kc+2*hw+1, col = col16
    v2f bih[2][3];   // x @ W_ih^T : K=8 -> 2 chunks, 3 N-tiles of 16 gate cols
#pragma unroll
    for (int kc = 0; kc < 2; kc++)
#pragma unroll
        for (int n = 0; n < 3; n++) {
            int cg = 16 * n + col16;
            int k0 = 4 * kc + 2 * hw;
            bih[kc][n].x = W_ih[cg * I_SZ + k0];
            bih[kc][n].y = W_ih[cg * I_SZ + k0 + 1];
        }
    v2f bhh[3][3];   // h @ W_hh^T : K=12 -> 3 chunks
#pragma unroll
    for (int kc = 0; kc < 3; kc++)
#pragma unroll
        for (int n = 0; n < 3; n++) {
            int cg = 16 * n + col16;
            int k0 = 4 * kc + 2 * hw;
            bhh[kc][n].x = W_hh[cg * H_SZ + k0];
            bhh[kc][n].y = W_hh[cg * H_SZ + k0 + 1];
        }
    v2f bfc[3];      // h @ W_fc^T : 4 valid cols, padded to 16 with zeros
#pragma unroll
    for (int kc = 0; kc < 3; kc++) {
        int k0 = 4 * kc + 2 * hw;
        float vx = 0.f, vy = 0.f;
        if (col16 < NA) { vx = W_fc[col16 * H_SZ + k0]; vy = W_fc[col16 * H_SZ + k0 + 1]; }
        bfc[kc].x = vx; bfc[kc].y = vy;
    }
    float biasv[3];
#pragma unroll
    for (int n = 0; n < 3; n++) {
        int cg = 16 * n + col16;
        biasv[n] = b_ih[cg] + b_hh[cg];
    }
    const float bfcv = (col16 < NA) ? b_fc[col16] : 0.f;
    bool useTanh[3];
#pragma unroll
    for (int n = 0; n < 3; n++) {
        int cg = 16 * n + col16;
        useTanh[n] = (cg >= 24 && cg < 36);   // 'g' gate -> tanh, others sigmoid
    }

    // ---- Init h, c = 0 in LDS (192 entries per wave, 6 per lane) ----
#pragma unroll
    for (int q = 0; q < 6; q++) {
        int p = lane + 32 * q;
        int r = p / H_SZ, j = p - r * H_SZ;
        hS[wid][r][j] = 0.f;
        cS[wid][r][j] = 0.f;
    }
    __syncthreads();

    // h A-fragments (A 16x4 f32: VGPR0 K=4kc+2*hw, VGPR1 K=4kc+2*hw+1, row M=col16)
    v2f ha[3];
#pragma unroll
    for (int kc = 0; kc < 3; kc++) { ha[kc].x = 0.f; ha[kc].y = 0.f; }

    const int rowA = col16;
    const float* xrow = x + (long)(b0 + rowA) * T_SZ * I_SZ;

#pragma unroll 1
    for (int t = 0; t < T_SZ; t++) {
        // ---- x A-fragments for this timestep: two float2 loads per lane ----
        const float* px = xrow + (long)t * I_SZ;
        v2f xa0, xa1;
        xa0.x = px[0 + 2 * hw]; xa0.y = px[1 + 2 * hw];
        xa1.x = px[4 + 2 * hw]; xa1.y = px[5 + 2 * hw];

        // prefetch x a few timesteps ahead (global_prefetch_b8; no LOADcnt cost)
        if (t + PF_T < T_SZ)
            __builtin_prefetch(xrow + (long)(t + PF_T) * I_SZ, 0, 3);

        // ---- gates[16,48] = bias + X*W_ih^T + H*W_hh^T via 15 WMMAs ----
        v8f acc[3];
#pragma unroll
        for (int n = 0; n < 3; n++) {
            v8f a;
#pragma unroll
            for (int e = 0; e < 8; e++) a[e] = biasv[n];
            a = __builtin_amdgcn_wmma_f32_16x16x4_f32(false, xa0,   false, bih[0][n], (short)0, a, false, false);
            a = __builtin_amdgcn_wmma_f32_16x16x4_f32(false, xa1,   false, bih[1][n], (short)0, a, false, false);
            a = __builtin_amdgcn_wmma_f32_16x16x4_f32(false, ha[0], false, bhh[0][n], (short)0, a, false, false);
            a = __builtin_amdgcn_wmma_f32_16x16x4_f32(false, ha[1], false, bhh[1][n], (short)0, a, false, false);
            a = __builtin_amdgcn_wmma_f32_16x16x4_f32(false, ha[2], false, bhh[2][n], (short)0, a, false, false);
            acc[n] = a;
        }

        // ---- activations (lane-local) then stage to LDS for regrouping ----
#pragma unroll
        for (int n = 0; n < 3; n++) {
#pragma unroll
            for (int e = 0; e < 8; e++) {
                float v = acc[n][e];
                float a = useTanh[n] ? fast_tanh(v) : fast_sigmoid(v);
                gatesS[wid][e + 8 * hw][16 * n + col16] = a;
            }
        }
        __syncthreads();

        // ---- pointwise state update: 6 (row, j) pairs per lane ----
#pragma unroll
        for (int q = 0; q < 6; q++) {
            int p = lane + 32 * q;
            int r = p / H_SZ, j = p - r * H_SZ;
            float ig = gatesS[wid][r][j];
            float fg = gatesS[wid][r][12 + j];
            float gg = gatesS[wid][r][24 + j];
            float og = gatesS[wid][r][36 + j];
            float cc = fg * cS[wid][r][j] + ig * gg;
            float hh = og * fast_tanh(cc);
            cS[wid][r][j] = cc;
            hS[wid][r][j] = hh;
        }
        __syncthreads();

        // ---- reload h as A-fragments for next step + FC head ----
#pragma unroll
        for (int kc = 0; kc < 3; kc++) {
            ha[kc].x = hS[wid][rowA][4 * kc + 2 * hw];
            ha[kc].y = hS[wid][rowA][4 * kc + 2 * hw + 1];
        }

        // ---- out[16,4] = h @ W_fc^T + b_fc via 3 WMMAs (cols 4..15 padded) ----
        v8f od;
#pragma unroll
        for (int e = 0; e < 8; e++) od[e] = bfcv;
        od = __builtin_amdgcn_wmma_f32_16x16x4_f32(false, ha[0], false, bfc[0], (short)0, od, false, false);
        od = __builtin_amdgcn_wmma_f32_16x16x4_f32(false, ha[1], false, bfc[1], (short)0, od, false, false);
        od = __builtin_amdgcn_wmma_f32_16x16x4_f32(false, ha[2], false, bfc[2], (short)0, od, false, false);

        if (col16 < NA) {
#pragma unroll
            for (int e = 0; e < 8; e++) {
                int b = b0 + e + 8 * hw;
                out[((long)b * T_SZ + t) * NA + col16] = od[e];
            }
        }
    }

    // ---- final hidden / cell state ----
#pragma unroll
    for (int q = 0; q < 6; q++) {
        int p = lane + 32 * q;
        int r = p / H_SZ, j = p - r * H_SZ;
        hn[(long)(b0 + r) * H_SZ + j] = hS[wid][r][j];
        cn[(long)(b0 + r) * H_SZ + j] = cS[wid][r][j];
    }
}

extern "C" void kernel_launch(void* const* d_in, const int* in_sizes, int n_in,
                              void* d_out, int out_size, void* d_ws, size_t ws_size,
                              hipStream_t stream) {
    (void)in_sizes; (void)n_in; (void)d_ws; (void)ws_size; (void)out_size;
    const float* x    = (const float*)d_in[0];
    const float* W_ih = (const float*)d_in[1];
    const float* W_hh = (const float*)d_in[2];
    const float* b_ih = (const float*)d_in[3];
    const float* b_hh = (const float*)d_in[4];
    const float* W_fc = (const float*)d_in[5];
    const float* b_fc = (const float*)d_in[6];

    float* out = (float*)d_out;                       // [B, T, 4]
    float* hn  = out + (size_t)B_SZ * T_SZ * NA;      // [1, B, 12]
    float* cn  = hn + (size_t)B_SZ * H_SZ;            // [1, B, 12]

    dim3 grid(B_SZ / (WPB * ROWS));                   // 128 blocks
    dim3 block(WPB * 32);                             // 2 waves of 32
    lstm_wmma_kernel<<<grid, block, 0, stream>>>(x, W_ih, W_hh, b_ih, b_hh,
                                                 W_fc, b_fc, out, hn, cn);
}